// GAT_60464549593286
// MI455X (gfx1250) — compile-verified
//
#include <hip/hip_runtime.h>
#include <math.h>

typedef __attribute__((ext_vector_type(2))) float v2f;
typedef __attribute__((ext_vector_type(8))) float v8f;

#define NEG_SLOPE 0.2f

// Order-preserving uint encoding of float for atomicMax-based segment max.
__device__ __forceinline__ unsigned enc_f32(float f) {
  unsigned u = __float_as_uint(f);
  return (u & 0x80000000u) ? ~u : (u | 0x80000000u);
}
__device__ __forceinline__ float dec_f32(unsigned u) {
  unsigned v = (u & 0x80000000u) ? (u ^ 0x80000000u) : ~u;
  return __uint_as_float(v);
}

__device__ __forceinline__ float leaky(float e) {
  return e > 0.f ? e : NEG_SLOPE * e;
}

// src/dst for edge id; ids >= E are the PyG-style appended self loops.
__device__ __forceinline__ void edge_sd(const long long* __restrict__ ei,
                                        int E, int eidx, int& s, int& d) {
  if (eidx < E) {
    s = (int)ei[eidx];
    d = (int)ei[(size_t)E + eidx];
  } else {
    s = eidx - E;
    d = s;
  }
}

// ---------------------------------------------------------------------------
// C[M,NC] = A[M,K] @ B[K,NC], fp32, V_WMMA_F32_16X16X4_F32.
// Block = 128 threads = 4 waves; each wave owns a 16x16 output tile.
// The 16xK A tile is staged in LDS once and shared by the 4 column tiles.
// No early-exit anywhere before the WMMA loop -> EXEC is all 1s (ISA req).
// Fragment layouts per ISA 7.12.2:
//   A (16x4 f32, 2 VGPRs): lane m = lane&15 is the row; half-wave (lane>>4)
//     selects K pair {0,1} vs {2,3}; VGPR0=K_even, VGPR1=K_odd.
//   B (4x16 f32, 2 VGPRs): symmetric, lane&15 is the column.
//   C/D (16x16 f32, 8 VGPRs): VGPR r -> row r (lanes 0-15) / r+8 (lanes 16-31),
//     column = lane&15.
// ---------------------------------------------------------------------------
template <int K, int NC>
__launch_bounds__(128)
__global__ void gemm_wmma_f32(const float* __restrict__ A,
                              const float* __restrict__ B,
                              float* __restrict__ C, int M) {
  __shared__ float As[16 * K];
  const int tid = threadIdx.x;
  const int wave = tid >> 5;
  const int lane = tid & 31;
  const int row0 = blockIdx.x * 16;
  const int col0 = (blockIdx.y * 4 + wave) * 16;

  // Cooperative float4 load of the contiguous 16xK A tile (rows clamped).
  const float4* Ag = (const float4*)(A + (size_t)row0 * K);
  float4* As4 = (float4*)As;
  #pragma unroll 4
  for (int i = tid; i < 16 * K / 4; i += 128) {
    const int r = (i * 4) / K;
    float4 v = make_float4(0.f, 0.f, 0.f, 0.f);
    if (row0 + r < M) v = Ag[i];
    As4[i] = v;
  }
  __syncthreads();

  const int m = lane & 15;          // A row / B,C column within tile
  const int kof = (lane >> 4) * 2;  // which K pair this half-wave holds

  v8f acc = {0.f, 0.f, 0.f, 0.f, 0.f, 0.f, 0.f, 0.f};
  #pragma unroll 4
  for (int k = 0; k < K; k += 4) {
    v2f a = *(const v2f*)(As + m * K + k + kof);
    v2f b;
    b.x = B[(size_t)(k + kof) * NC + col0 + m];
    b.y = B[(size_t)(k + kof + 1) * NC + col0 + m];
    acc = __builtin_amdgcn_wmma_f32_16x16x4_f32(false, a, false, b,
                                                (short)0, acc, false, false);
  }

  const int rbase = (lane >> 4) * 8;
  #pragma unroll
  for (int r = 0; r < 8; ++r) {
    const int gr = row0 + rbase + r;
    if (gr < M) C[(size_t)gr * NC + col0 + m] = acc[r];
  }
}

// ---------------------------------------------------------------------------
// Per node (wave each): a_src[n]=dot(h[n],att_s), a_dst[n]=dot(h[n],att_d),
// init emax(enc)=0, denom=0, out[n,:] = bias (atomic accumulation target).
// ---------------------------------------------------------------------------
template <int C>
__launch_bounds__(256)
__global__ void attn_prep(const float* __restrict__ h,
                          const float* __restrict__ att_s,
                          const float* __restrict__ att_d,
                          const float* __restrict__ bias,
                          float* __restrict__ a_src, float* __restrict__ a_dst,
                          unsigned* __restrict__ emax,
                          float* __restrict__ denom,
                          float* __restrict__ out, int N) {
  const int lane = threadIdx.x & 31;
  const int node = blockIdx.x * 8 + (threadIdx.x >> 5);
  if (node >= N) return;

  constexpr int V = C / 32;
  float vals[V];
  const float* hr = h + (size_t)node * C + lane * V;
  if constexpr (V == 4) {
    *(float4*)vals = *(const float4*)hr;
  } else {
    *(float2*)vals = *(const float2*)hr;
  }

  float ps = 0.f, pd = 0.f;
  #pragma unroll
  for (int i = 0; i < V; ++i) {
    const int c = lane * V + i;
    ps += vals[i] * att_s[c];
    pd += vals[i] * att_d[c];
    out[(size_t)node * C + c] = bias[c];
  }
  #pragma unroll
  for (int off = 16; off > 0; off >>= 1) {
    ps += __shfl_xor(ps, off, 32);
    pd += __shfl_xor(pd, off, 32);
  }
  if (lane == 0) {
    a_src[node] = ps;
    a_dst[node] = pd;
    emax[node] = 0u;  // below enc(-inf); every node has a self loop anyway
    denom[node] = 0.f;
  }
}

// Pass A: segment max of leaky(e) over dst via encoded uint atomicMax.
__launch_bounds__(256)
__global__ void edge_max(const long long* __restrict__ ei, int E, int N,
                         const float* __restrict__ a_src,
                         const float* __restrict__ a_dst,
                         unsigned* __restrict__ emax) {
  const int i = blockIdx.x * blockDim.x + threadIdx.x;
  if (i >= E + N) return;
  int s, d;
  edge_sd(ei, E, i, s, d);
  const float e = leaky(a_src[s] + a_dst[d]);
  atomicMax(&emax[d], enc_f32(e));
}

// Pass B: denom[dst] += exp(e - emax[dst]).
__launch_bounds__(256)
__global__ void edge_expsum(const long long* __restrict__ ei, int E, int N,
                            const float* __restrict__ a_src,
                            const float* __restrict__ a_dst,
                            const unsigned* __restrict__ emax,
                            float* __restrict__ denom) {
  const int i = blockIdx.x * blockDim.x + threadIdx.x;
  if (i >= E + N) return;
  int s, d;
  edge_sd(ei, E, i, s, d);
  const float e = leaky(a_src[s] + a_dst[d]);
  atomicAdd(&denom[d], expf(e - dec_f32(emax[d])));
}

// Pass C: out[dst,:] += alpha * h[src,:]  (wave per edge, vector loads,
// per-channel global_atomic_add_f32).
template <int C>
__launch_bounds__(256)
__global__ void edge_aggregate(const long long* __restrict__ ei, int E, int N,
                               const float* __restrict__ a_src,
                               const float* __restrict__ a_dst,
                               const unsigned* __restrict__ emax,
                               const float* __restrict__ denom,
                               const float* __restrict__ h,
                               float* __restrict__ out) {
  const int lane = threadIdx.x & 31;
  const int eidx = blockIdx.x * 8 + (threadIdx.x >> 5);
  if (eidx >= E + N) return;
  int s, d;
  edge_sd(ei, E, eidx, s, d);
  const float e = leaky(a_src[s] + a_dst[d]);
  const float alpha = expf(e - dec_f32(emax[d])) / denom[d];

  constexpr int V = C / 32;
  float vals[V];
  const float* hs = h + (size_t)s * C + lane * V;
  if constexpr (V == 4) {
    *(float4*)vals = *(const float4*)hs;
  } else {
    *(float2*)vals = *(const float2*)hs;
  }
  float* od = out + (size_t)d * C + lane * V;
  #pragma unroll
  for (int i = 0; i < V; ++i) atomicAdd(od + i, vals[i] * alpha);
}

__launch_bounds__(256)
__global__ void relu_ip(float* __restrict__ x, long long n4) {
  const long long i = (long long)blockIdx.x * blockDim.x + threadIdx.x;
  if (i >= n4) return;
  float4 v = ((float4*)x)[i];
  v.x = fmaxf(v.x, 0.f);
  v.y = fmaxf(v.y, 0.f);
  v.z = fmaxf(v.z, 0.f);
  v.w = fmaxf(v.w, 0.f);
  ((float4*)x)[i] = v;
}

// In-place log_softmax over 64 channels, one wave per node (2 vals / lane).
__launch_bounds__(256)
__global__ void log_softmax64(float* __restrict__ x, int N) {
  const int lane = threadIdx.x & 31;
  const int node = blockIdx.x * 8 + (threadIdx.x >> 5);
  if (node >= N) return;
  float* row = x + (size_t)node * 64;
  float2 v = ((float2*)row)[lane];
  float mx = fmaxf(v.x, v.y);
  #pragma unroll
  for (int off = 16; off > 0; off >>= 1) mx = fmaxf(mx, __shfl_xor(mx, off, 32));
  float sum = expf(v.x - mx) + expf(v.y - mx);
  #pragma unroll
  for (int off = 16; off > 0; off >>= 1) sum += __shfl_xor(sum, off, 32);
  const float lse = mx + logf(sum);
  v.x -= lse;
  v.y -= lse;
  ((float2*)row)[lane] = v;
}

extern "C" void kernel_launch(void* const* d_in, const int* in_sizes, int n_in,
                              void* d_out, int out_size, void* d_ws,
                              size_t ws_size, hipStream_t stream) {
  const float* x = (const float*)d_in[0];
  const long long* ei = (const long long*)d_in[1];  // [2, E] int64
  const float* W1 = (const float*)d_in[2];
  const float* as1 = (const float*)d_in[3];
  const float* ad1 = (const float*)d_in[4];
  const float* b1 = (const float*)d_in[5];
  const float* W2 = (const float*)d_in[6];
  const float* as2 = (const float*)d_in[7];
  const float* ad2 = (const float*)d_in[8];
  const float* b2 = (const float*)d_in[9];

  constexpr int IN_C = 256, HID = 128, OUT_C = 64;
  const int N = in_sizes[0] / IN_C;  // 50000
  const int E = in_sizes[1] / 2;     // 800000
  const int ET = E + N;              // edges + self loops

  // Workspace carve-up (hipMalloc base is >=256B aligned; all chunks 16B mult)
  char* ws = (char*)d_ws;
  float* h1 = (float*)ws;      ws += (size_t)N * HID * sizeof(float);
  float* out1 = (float*)ws;    ws += (size_t)N * HID * sizeof(float);
  float* h2 = (float*)ws;      ws += (size_t)N * OUT_C * sizeof(float);
  float* aS = (float*)ws;      ws += (size_t)N * sizeof(float);
  float* aD = (float*)ws;      ws += (size_t)N * sizeof(float);
  unsigned* eM = (unsigned*)ws; ws += (size_t)N * sizeof(unsigned);
  float* den = (float*)ws;     ws += (size_t)N * sizeof(float);
  float* outF = (float*)d_out;  // layer-2 accumulator == final output buffer

  const dim3 b128(128), b256(256);
  const int nodeBlocks = (N + 7) / 8;
  const int edgeBlocks1 = (ET + 255) / 256;
  const int edgeBlocks8 = (ET + 7) / 8;

  // ---------------- Layer 1 ----------------
  gemm_wmma_f32<IN_C, HID>
      <<<dim3((N + 15) / 16, HID / 64), b128, 0, stream>>>(x, W1, h1, N);
  attn_prep<HID><<<nodeBlocks, b256, 0, stream>>>(h1, as1, ad1, b1, aS, aD,
                                                  eM, den, out1, N);
  edge_max<<<edgeBlocks1, b256, 0, stream>>>(ei, E, N, aS, aD, eM);
  edge_expsum<<<edgeBlocks1, b256, 0, stream>>>(ei, E, N, aS, aD, eM, den);
  edge_aggregate<HID>
      <<<edgeBlocks8, b256, 0, stream>>>(ei, E, N, aS, aD, eM, den, h1, out1);
  relu_ip<<<(int)(((long long)N * HID / 4 + 255) / 256), b256, 0, stream>>>(
      out1, (long long)N * HID / 4);

  // ---------------- Layer 2 ----------------
  gemm_wmma_f32<HID, OUT_C>
      <<<dim3((N + 15) / 16, OUT_C / 64), b128, 0, stream>>>(out1, W2, h2, N);
  attn_prep<OUT_C><<<nodeBlocks, b256, 0, stream>>>(h2, as2, ad2, b2, aS, aD,
                                                    eM, den, outF, N);
  edge_max<<<edgeBlocks1, b256, 0, stream>>>(ei, E, N, aS, aD, eM);
  edge_expsum<<<edgeBlocks1, b256, 0, stream>>>(ei, E, N, aS, aD, eM, den);
  edge_aggregate<OUT_C>
      <<<edgeBlocks8, b256, 0, stream>>>(ei, E, N, aS, aD, eM, den, h2, outF);
  log_softmax64<<<nodeBlocks, b256, 0, stream>>>(outF, N);
}